// SAGE_20117626814731
// MI455X (gfx1250) — compile-verified
//
#include <hip/hip_runtime.h>
#include <math.h>

// GraphSAGE 2-layer forward for MI455X (gfx1250, wave32).
// Memory-bound in the edge scatter; GEMMs use V_WMMA_F32_16X16X4_F32 to keep
// full f32 precision (the matmul FLOPs are negligible vs 23.3 TB/s HBM).

#define DIN  64
#define DHID 128
#define DOUT 40
#define NPAD 48   // DOUT padded to 3 WMMA n-tiles

typedef __attribute__((ext_vector_type(2))) float v2f;
typedef __attribute__((ext_vector_type(8))) float v8f;

__device__ __forceinline__ v8f wmma_f32(v2f a, v2f b, v8f c) {
  // D = A(16x4 f32) x B(4x16 f32) + C(16x16 f32)
  return __builtin_amdgcn_wmma_f32_16x16x4_f32(
      /*neg_a=*/false, a, /*neg_b=*/false, b,
      /*c_mod=*/(short)0, c, /*reuse_a=*/false, /*reuse_b=*/false);
}

// ---------------------------------------------------------------- zero scratch
__global__ void k_zero(float4* __restrict__ ws, long n4a, long skip4, long n4tot) {
  long i = (long)blockIdx.x * blockDim.x + threadIdx.x;
  if (i >= n4tot) return;
  long j = (i < n4a) ? i : i + skip4;   // skip over the 'h' region (overwritten)
  ws[j] = make_float4(0.f, 0.f, 0.f, 0.f);
}

// ---------------------------------------------------------------- degree count
__global__ void k_count(const int* __restrict__ dst, float* __restrict__ cnt, int E) {
  int e = blockIdx.x * blockDim.x + threadIdx.x;
  if (e < E) atomicAdd(&cnt[dst[e]], 1.0f);
}

// ------------------------------------------------------- edge scatter-add (sum)
template <int D>
__global__ void k_scatter(const int* __restrict__ src, const int* __restrict__ dst,
                          const float* __restrict__ feat, float* __restrict__ agg,
                          int E) {
  const int per = D / 4;                 // float4 chunks per edge
  long gid = (long)blockIdx.x * blockDim.x + threadIdx.x;
  int  e   = (int)(gid / per);
  if (e >= E) return;
  int  q   = (int)(gid % per) * 4;
  int  s   = src[e], d = dst[e];
  float4 v = *(const float4*)(feat + (size_t)s * D + q);
  float* a = agg + (size_t)d * D + q;
  atomicAdd(a + 0, v.x);
  atomicAdd(a + 1, v.y);
  atomicAdd(a + 2, v.z);
  atomicAdd(a + 3, v.w);
}

// -------------------------------------------- layer 1: (agg/cnt)@W1l' + x@W1r'
// One wave -> 16 rows x 128 cols (8 n-tiles). 8 waves/block, weights in LDS.
__global__ void __launch_bounds__(256)
k_sage1(const float* __restrict__ x, const float* __restrict__ agg,
        const float* __restrict__ cnt, const float* __restrict__ Wl,
        const float* __restrict__ bias, const float* __restrict__ Wr,
        float* __restrict__ h, int N) {
  __shared__ float sWl[DHID * DIN];   // 32 KB
  __shared__ float sWr[DHID * DIN];   // 32 KB
  for (int i = threadIdx.x; i < DHID * DIN / 4; i += 256) {
    ((float4*)sWl)[i] = ((const float4*)Wl)[i];
    ((float4*)sWr)[i] = ((const float4*)Wr)[i];
  }
  __syncthreads();

  const int lane = threadIdx.x & 31;
  const int wave = threadIdx.x >> 5;
  const int m    = lane & 15;
  const int half = lane >> 4;                       // 0 or 1 (K pair select)
  const int n0   = lane & 15;                       // column within n-tile
  const int row0 = (blockIdx.x * 8 + wave) * 16;
  int rowA = row0 + m; if (rowA > N - 1) rowA = N - 1;   // clamp loads, EXEC stays full
  const float invc = 1.0f / fmaxf(cnt[rowA], 1.0f);

  v8f acc[8] = {};

  // pass 1: mean-aggregated neighbors x W1l^T
#pragma unroll
  for (int k0 = 0; k0 < DIN; k0 += 4) {
    const int kk = k0 + 2 * half;
    float2 av = *(const float2*)(agg + (size_t)rowA * DIN + kk);
    v2f a; a.x = av.x * invc; a.y = av.y * invc;
#pragma unroll
    for (int t = 0; t < 8; ++t) {
      float2 bv = *(const float2*)(sWl + (t * 16 + n0) * DIN + kk);  // B = W^T
      v2f b; b.x = bv.x; b.y = bv.y;
      acc[t] = wmma_f32(a, b, acc[t]);
    }
  }
  // pass 2: root features x W1r^T (accumulate into same C)
#pragma unroll
  for (int k0 = 0; k0 < DIN; k0 += 4) {
    const int kk = k0 + 2 * half;
    float2 av = *(const float2*)(x + (size_t)rowA * DIN + kk);
    v2f a; a.x = av.x; a.y = av.y;
#pragma unroll
    for (int t = 0; t < 8; ++t) {
      float2 bv = *(const float2*)(sWr + (t * 16 + n0) * DIN + kk);
      v2f b; b.x = bv.x; b.y = bv.y;
      acc[t] = wmma_f32(a, b, acc[t]);
    }
  }
  // epilogue: + bias, ReLU, store
#pragma unroll
  for (int t = 0; t < 8; ++t) {
    const int   n = t * 16 + n0;
    const float bb = bias[n];
#pragma unroll
    for (int r = 0; r < 8; ++r) {
      const int row = row0 + r + 8 * half;          // C layout: M = vgpr + 8*(lane/16)
      if (row < N) {
        float v = acc[t][r] + bb;
        h[(size_t)row * DHID + n] = fmaxf(v, 0.0f);
      }
    }
  }
}

// --------------------------------------- layer 2: (agg/cnt)@W2l' + h@W2r' + b2
__global__ void __launch_bounds__(256)
k_sage2(const float* __restrict__ hin, const float* __restrict__ agg,
        const float* __restrict__ cnt, const float* __restrict__ Wl,
        const float* __restrict__ bias, const float* __restrict__ Wr,
        float* __restrict__ out, int N) {
  __shared__ float sWl[NPAD * DHID];  // 24 KB (rows 40..47 zero)
  __shared__ float sWr[NPAD * DHID];  // 24 KB
  for (int i = threadIdx.x; i < NPAD * DHID / 4; i += 256) {
    ((float4*)sWl)[i] = make_float4(0.f, 0.f, 0.f, 0.f);
    ((float4*)sWr)[i] = make_float4(0.f, 0.f, 0.f, 0.f);
  }
  __syncthreads();
  for (int i = threadIdx.x; i < DOUT * DHID / 4; i += 256) {   // rows 0..39 contiguous
    ((float4*)sWl)[i] = ((const float4*)Wl)[i];
    ((float4*)sWr)[i] = ((const float4*)Wr)[i];
  }
  __syncthreads();

  const int lane = threadIdx.x & 31;
  const int wave = threadIdx.x >> 5;
  const int m    = lane & 15;
  const int half = lane >> 4;
  const int n0   = lane & 15;
  const int row0 = (blockIdx.x * 8 + wave) * 16;
  int rowA = row0 + m; if (rowA > N - 1) rowA = N - 1;
  const float invc = 1.0f / fmaxf(cnt[rowA], 1.0f);

  v8f acc[3] = {};

#pragma unroll 4
  for (int k0 = 0; k0 < DHID; k0 += 4) {
    const int kk = k0 + 2 * half;
    float2 av = *(const float2*)(agg + (size_t)rowA * DHID + kk);
    v2f a; a.x = av.x * invc; a.y = av.y * invc;
#pragma unroll
    for (int t = 0; t < 3; ++t) {
      float2 bv = *(const float2*)(sWl + (t * 16 + n0) * DHID + kk);
      v2f b; b.x = bv.x; b.y = bv.y;
      acc[t] = wmma_f32(a, b, acc[t]);
    }
  }
#pragma unroll 4
  for (int k0 = 0; k0 < DHID; k0 += 4) {
    const int kk = k0 + 2 * half;
    float2 av = *(const float2*)(hin + (size_t)rowA * DHID + kk);
    v2f a; a.x = av.x; a.y = av.y;
#pragma unroll
    for (int t = 0; t < 3; ++t) {
      float2 bv = *(const float2*)(sWr + (t * 16 + n0) * DHID + kk);
      v2f b; b.x = bv.x; b.y = bv.y;
      acc[t] = wmma_f32(a, b, acc[t]);
    }
  }
#pragma unroll
  for (int t = 0; t < 3; ++t) {
    const int n = t * 16 + n0;
    if (n < DOUT) {
      const float bb = bias[n];
#pragma unroll
      for (int r = 0; r < 8; ++r) {
        const int row = row0 + r + 8 * half;
        if (row < N) out[(size_t)row * DOUT + n] = acc[t][r] + bb;
      }
    }
  }
}

// ------------------------------------------------------- in-place log_softmax
__global__ void k_logsoftmax(float* __restrict__ out, int N) {
  int row = blockIdx.x * blockDim.x + threadIdx.x;
  if (row >= N) return;
  float* p = out + (size_t)row * DOUT;
  float v[DOUT];
  float mx = -3.402823466e38f;
#pragma unroll
  for (int j = 0; j < DOUT; ++j) { v[j] = p[j]; mx = fmaxf(mx, v[j]); }
  float s = 0.f;
#pragma unroll
  for (int j = 0; j < DOUT; ++j) s += expf(v[j] - mx);
  float lse = logf(s) + mx;
#pragma unroll
  for (int j = 0; j < DOUT; ++j) p[j] = v[j] - lse;
}

// ------------------------------------------------------------------- launcher
extern "C" void kernel_launch(void* const* d_in, const int* in_sizes, int n_in,
                              void* d_out, int out_size, void* d_ws, size_t ws_size,
                              hipStream_t stream) {
  const float* x   = (const float*)d_in[0];
  const int*   ei  = (const int*)d_in[1];      // (2, E) int32 (JAX x64 off)
  const float* W1l = (const float*)d_in[2];
  const float* b1  = (const float*)d_in[3];
  const float* W1r = (const float*)d_in[4];
  const float* W2l = (const float*)d_in[5];
  const float* b2  = (const float*)d_in[6];
  const float* W2r = (const float*)d_in[7];
  float* out = (float*)d_out;

  const int N = in_sizes[0] / DIN;
  const int E = in_sizes[1] / 2;
  const int* src = ei;
  const int* dst = ei + E;

  // workspace layout (floats): cnt[N] | agg1[N*64] | h[N*128] | agg2[N*128]
  float* ws   = (float*)d_ws;
  float* cnt  = ws;
  float* agg1 = ws + (size_t)N;
  float* h    = ws + (size_t)N * 65;
  float* agg2 = ws + (size_t)N * 193;

  // zero cnt+agg1 and agg2 every call (h is fully overwritten)
  long n4a = (long)N * 65 / 4, skip4 = (long)N * 128 / 4, n4tot = (long)N * 193 / 4;
  k_zero<<<(unsigned)((n4tot + 255) / 256), 256, 0, stream>>>((float4*)ws, n4a, skip4, n4tot);

  k_count<<<(E + 255) / 256, 256, 0, stream>>>(dst, cnt, E);

  { long T = (long)E * (DIN / 4);
    k_scatter<DIN><<<(unsigned)((T + 255) / 256), 256, 0, stream>>>(src, dst, x, agg1, E); }

  int gemm_blocks = (N + 127) / 128;
  k_sage1<<<gemm_blocks, 256, 0, stream>>>(x, agg1, cnt, W1l, b1, W1r, h, N);

  { long T = (long)E * (DHID / 4);
    k_scatter<DHID><<<(unsigned)((T + 255) / 256), 256, 0, stream>>>(src, dst, h, agg2, E); }

  k_sage2<<<gemm_blocks, 256, 0, stream>>>(h, agg2, cnt, W2l, b2, W2r, out, N);

  k_logsoftmax<<<(N + 255) / 256, 256, 0, stream>>>(out, N);
}